// MultiEchoNeighborBlock_34428457845311
// MI455X (gfx1250) — compile-verified
//
#include <hip/hip_runtime.h>
#include <hip/hip_bf16.h>

// Problem constants (from reference)
#define N_ECHOES 2
#define SEARCH   7
#define PADR     3
#define KNN      9
#define STEM     32
#define NFEAT    (KNN * N_ECHOES + N_ECHOES)   // 20
#define HH       64
#define WWID     2048
#define HWD      (HH * WWID)                   // 131072
#define NBATCH   4

typedef __attribute__((ext_vector_type(2))) float v2f;
typedef __attribute__((ext_vector_type(8))) float v8f;

// Insert (d, r) into a 9-entry ascending priority queue held in registers.
// Strict '<' keeps earlier (lower-index) entries ahead on ties, matching
// jax.lax.top_k's first-occurrence tie-break on -dist.
__device__ __forceinline__ void insert9(float d, float r, float kd[KNN], float kr[KNN]) {
    bool c[KNN];
#pragma unroll
    for (int i = 0; i < KNN; ++i) c[i] = d < kd[i];
#pragma unroll
    for (int i = KNN - 1; i >= 1; --i) {
        kd[i] = c[i] ? (c[i - 1] ? kd[i - 1] : d) : kd[i];
        kr[i] = c[i] ? (c[i - 1] ? kr[i - 1] : r) : kr[i];
    }
    kd[0] = c[0] ? d : kd[0];
    kr[0] = c[0] ? r : kr[0];
}

__global__ __launch_bounds__(256) void MultiEchoNeighborBlock_kernel(
    const float* __restrict__ x,     // (4, 8, 64, 2048)
    const float* __restrict__ w,     // (1, 32, 20) -> w[k*20 + c]
    float* __restrict__ out)         // (4, 32, 64, 2048)
{
    __shared__ float wlds[STEM * NFEAT];      // 640 f32
    __shared__ float feats[256 * NFEAT];      // 256 pixels x 20 feats

    const int tid = threadIdx.x;

    // Stage the 32x20 weight matrix in LDS.
    for (int i = tid; i < STEM * NFEAT; i += 256) wlds[i] = w[i];

    // 512 blocks per batch image (HW / 256).
    const int b       = blockIdx.x >> 9;
    const int pixbase = (blockIdx.x & 511) << 8;
    const int pix     = pixbase + tid;
    const int h       = pix >> 11;          // W = 2048
    const int wcol    = pix & (WWID - 1);

    const float* __restrict__ xb = x + (size_t)b * 8 * HWD;

    // Center values: ranges of both echoes + xyz points of both echoes.
    const float r0  = xb[pix];
    const float r1  = xb[1 * HWD + pix];
    const float p0x = xb[2 * HWD + pix], p0y = xb[3 * HWD + pix], p0z = xb[4 * HWD + pix];
    const float p1x = xb[5 * HWD + pix], p1y = xb[6 * HWD + pix], p1z = xb[7 * HWD + pix];

    float kd0[KNN], kr0[KNN], kd1[KNN], kr1[KNN];
#pragma unroll
    for (int i = 0; i < KNN; ++i) {
        kd0[i] = __builtin_inff(); kr0[i] = 0.f;
        kd1[i] = __builtin_inff(); kr1[i] = 0.f;
    }

    // One pass over the 7x7 neighborhood of echo-0 (range + xyz), feeding both
    // echo queues. Out-of-bounds neighbors behave as zero (unfold zero-padding).
#pragma unroll
    for (int j = 0; j < SEARCH * SEARCH; ++j) {
        const int dy = j / SEARCH - PADR;
        const int dx = j % SEARCH - PADR;
        const int hh = h + dy, wc = wcol + dx;
        const bool inb = (hh >= 0) & (hh < HH) & (wc >= 0) & (wc < WWID);
        const int hcl = min(max(hh, 0), HH - 1);
        const int wcl = min(max(wc, 0), WWID - 1);
        const int off = hcl * WWID + wcl;
        const float m = inb ? 1.0f : 0.0f;
        const float qr = xb[off] * m;
        const float qx = xb[2 * HWD + off] * m;
        const float qy = xb[3 * HWD + off] * m;
        const float qz = xb[4 * HWD + off] * m;

        float ax = p0x - qx, ay = p0y - qy, az = p0z - qz;
        const float d0 = ax * ax + ay * ay + az * az;   // sqrt is order-preserving: skip it
        float bx = p1x - qx, by = p1y - qy, bz = p1z - qz;
        const float d1 = bx * bx + by * by + bz * bz;

        insert9(d0, qr, kd0, kr0);
        insert9(d1, qr, kd1, kr1);
    }

    // feats layout per pixel: [knn0(9), r0, knn1(9), r1]
    float* f = &feats[tid * NFEAT];
#pragma unroll
    for (int i = 0; i < KNN; ++i) f[i] = kr0[i];
    f[KNN] = r0;
#pragma unroll
    for (int i = 0; i < KNN; ++i) f[KNN + 1 + i] = kr1[i];
    f[2 * KNN + 1] = r1;

    __syncthreads();

    // ---- WMMA phase: out[k, n] = sum_c W[k,c] * feats[c,n], fp32 16x16x4 ----
    // A (16x4): lanes 0-15 hold K={0,1}, lanes 16-31 hold K={2,3}, row M = lane%16.
    // B (4x16): lanes 0-15 hold K={0,1}, lanes 16-31 hold K={2,3}, col N = lane%16.
    // C/D (16x16): VGPR v -> row v + 8*(lane>=16), col lane%16.
    const int lane  = tid & 31;
    const int wave  = tid >> 5;
    const int ncol  = lane & 15;
    const int khalf = (lane >> 4) << 1;        // 0 or 2
    const int rhalf = (lane >> 4) << 3;        // 0 or 8 (D row offset)

    float* __restrict__ ob = out + (size_t)b * STEM * HWD;

#pragma unroll
    for (int t = 0; t < 2; ++t) {              // two 16-pixel tiles per wave
        const float* fp = &feats[(wave * 32 + t * 16 + ncol) * NFEAT];
        const int npix = pixbase + wave * 32 + t * 16 + ncol;
#pragma unroll
        for (int mt = 0; mt < 2; ++mt) {       // output channels 0-15 / 16-31
            const float* wp = &wlds[(mt * 16 + ncol) * NFEAT];
            v8f acc = {};
#pragma unroll
            for (int kc = 0; kc < 5; ++kc) {   // 20 = 5 chunks of K=4
                v2f a, bb;
                a.x  = wp[4 * kc + khalf];
                a.y  = wp[4 * kc + khalf + 1];
                bb.x = fp[4 * kc + khalf];
                bb.y = fp[4 * kc + khalf + 1];
                acc = __builtin_amdgcn_wmma_f32_16x16x4_f32(
                    /*neg_a=*/false, a, /*neg_b=*/false, bb,
                    /*c_mod=*/(short)0, acc, /*reuse_a=*/false, /*reuse_b=*/false);
            }
#pragma unroll
            for (int v = 0; v < 8; ++v) {
                const int krow = mt * 16 + rhalf + v;
                float val = acc[v];
                val = val > 0.f ? val : 0.01f * val;   // leaky_relu(0.01)
                ob[(size_t)krow * HWD + npix] = val;
            }
        }
    }
}

extern "C" void kernel_launch(void* const* d_in, const int* in_sizes, int n_in,
                              void* d_out, int out_size, void* d_ws, size_t ws_size,
                              hipStream_t stream) {
    (void)in_sizes; (void)n_in; (void)out_size; (void)d_ws; (void)ws_size;
    const float* x = (const float*)d_in[0];          // (4, 8, 64, 2048) f32
    const float* w = (const float*)d_in[1];          // (1, 32, 20) f32
    float* out = (float*)d_out;                      // (4, 32, 64, 2048) f32

    const int blocks = NBATCH * (HWD / 256);         // 2048 blocks of 256 threads
    MultiEchoNeighborBlock_kernel<<<blocks, 256, 0, stream>>>(x, w, out);
}